// CellularAutomataModel_70995809403525
// MI455X (gfx1250) — compile-verified
//
#include <hip/hip_runtime.h>
#include <math.h>

// -------------------------------------------------------------------------
// Fire-spread cellular automaton, MI455X (gfx1250).
// Memory-bound (~470 MB @ 23.3 TB/s ~= 20 us floor). Strategy:
//   * stage elev + fire halo tiles into LDS via GLOBAL_LOAD_ASYNC_TO_LDS_B32
//     (CDNA5 async path, ASYNCcnt), zero-fill out-of-image halo,
//   * coalesced float4 loads for the 4 elementwise channels, issued before
//     the async wait so they overlap the staging,
//   * exact trig identities: tan(atan(g)) = g;
//     cos(mod(270-wd,360)deg) = -sin(wd deg), sin(...) = -cos(wd deg);
//     max over dirs folded through the monotone clamp -> one fma+clamp.
// -------------------------------------------------------------------------

namespace {
constexpr int kH = 1024;
constexpr int kW = 1024;
constexpr int kC = 6;
constexpr int TW = 128;          // tile width  (interior)
constexpr int TH = 16;           // tile height (interior)
constexpr int HALO_W = TW + 2;   // 130
constexpr int HALO_H = TH + 2;   // 18
constexpr int LDSW = 132;        // padded LDS row stride (floats)
constexpr int NHALO = HALO_W * HALO_H;  // 2340
constexpr int NTHREADS = 256;    // 8 wave32s
constexpr float kPBurn  = 0.58f;
constexpr float kWWind  = 0.045f;
constexpr float kWSlope = 0.078f;
constexpr float kR2 = 0.70710678118654752440f;
constexpr float kDeg2Rad = 0.017453292519943295f;
} // namespace

__global__ __launch_bounds__(NTHREADS) void fire_spread_kernel(
    const float* __restrict__ x, float* __restrict__ out) {
  __shared__ float sE[HALO_H * LDSW];   // elev halo tile
  __shared__ float sF[HALO_H * LDSW];   // fire halo tile

  const int tid = threadIdx.x;
  const int c0  = blockIdx.x * TW;
  const int r0  = blockIdx.y * TH;
  const size_t CS = (size_t)kH * kW;
  const float* __restrict__ xb = x + (size_t)blockIdx.z * kC * CS;
  const float* __restrict__ xE = xb;            // CH_ELEV = 0
  const float* __restrict__ xF = xb + 5 * CS;   // CH_FIRE = 5

  // ---- stage halo tiles into LDS via CDNA5 async-to-LDS loads ------------
  for (int idx = tid; idx < NHALO; idx += NTHREADS) {
    const int lrr = idx / HALO_W;
    const int lcc = idx - lrr * HALO_W;
    const int gr  = r0 - 1 + lrr;
    const int gc  = c0 - 1 + lcc;
    const int l   = lrr * LDSW + lcc;
    if (gr >= 0 && gr < kH && gc >= 0 && gc < kW) {
      const size_t g = (size_t)gr * kW + gc;
      const unsigned ldsE = (unsigned)(unsigned long long)(const void*)&sE[l];
      const unsigned ldsF = (unsigned)(unsigned long long)(const void*)&sF[l];
      asm volatile("global_load_async_to_lds_b32 %0, %1, off"
                   :: "v"(ldsE), "v"(xE + g) : "memory");
      asm volatile("global_load_async_to_lds_b32 %0, %1, off"
                   :: "v"(ldsF), "v"(xF + g) : "memory");
    } else {
      // zero padding outside the image (disjoint LDS addresses from the
      // async writes, so no ordering hazard)
      sE[l] = 0.0f;
      sF[l] = 0.0f;
    }
  }

  // ---- each thread computes 8 consecutive pixels of the 16x128 tile ------
  const int p0 = tid * 8;
  const int lr = p0 >> 7;          // tile row 0..15
  const int lc = p0 & (TW - 1);    // tile col, multiple of 8
  const size_t grow = (size_t)(r0 + lr) * kW + (c0 + lc);

  // Elementwise channels: issue these global loads BEFORE waiting on the
  // async staging so the latencies overlap.
  float wsv[8], wdv[8], humv[8], ndvv[8];
  {
    const float4* p;
    float4 a, b;
    p = (const float4*)(xb + 1 * CS + grow);  a = p[0]; b = p[1];   // CH_WS
    wsv[0]=a.x; wsv[1]=a.y; wsv[2]=a.z; wsv[3]=a.w;
    wsv[4]=b.x; wsv[5]=b.y; wsv[6]=b.z; wsv[7]=b.w;
    p = (const float4*)(xb + 2 * CS + grow);  a = p[0]; b = p[1];   // CH_WD
    wdv[0]=a.x; wdv[1]=a.y; wdv[2]=a.z; wdv[3]=a.w;
    wdv[4]=b.x; wdv[5]=b.y; wdv[6]=b.z; wdv[7]=b.w;
    p = (const float4*)(xb + 3 * CS + grow);  a = p[0]; b = p[1];   // CH_HUM
    humv[0]=a.x; humv[1]=a.y; humv[2]=a.z; humv[3]=a.w;
    humv[4]=b.x; humv[5]=b.y; humv[6]=b.z; humv[7]=b.w;
    p = (const float4*)(xb + 4 * CS + grow);  a = p[0]; b = p[1];   // CH_NDVI
    ndvv[0]=a.x; ndvv[1]=a.y; ndvv[2]=a.z; ndvv[3]=a.w;
    ndvv[4]=b.x; ndvv[5]=b.y; ndvv[6]=b.z; ndvv[7]=b.w;
  }

  asm volatile("s_wait_asynccnt 0x0" ::: "memory");
  __syncthreads();

  float res[8];
#pragma unroll
  for (int j = 0; j < 8; ++j) {
    const int lR = lr + 1;       // LDS row of center
    const int lC = lc + 1 + j;   // LDS col of center

    // Sobel (cross-correlation, /8) on elev from LDS
    const float* e0 = &sE[(lR - 1) * LDSW + (lC - 1)];
    const float* e1 = &sE[(lR    ) * LDSW + (lC - 1)];
    const float* e2 = &sE[(lR + 1) * LDSW + (lC - 1)];
    const float a00 = e0[0], a01 = e0[1], a02 = e0[2];
    const float a10 = e1[0],             a12 = e1[2];
    const float a20 = e2[0], a21 = e2[1], a22 = e2[2];
    const float dx = ((a02 - a00) + 2.0f * (a12 - a10) + (a22 - a20)) * 0.125f;
    const float dy = ((a20 - a00) + 2.0f * (a21 - a01) + (a22 - a02)) * 0.125f;
    const float gmag = __builtin_amdgcn_sqrtf(fmaf(dx, dx, fmaf(dy, dy, 1e-8f)));

    // tan(atan(g) deg->rad round-trip) == g
    const float slope_eff = fmaf(kWSlope, gmag, 1.0f);
    const float moist = fminf(fmaxf(fmaf(-400.0f, humv[j], 1.0f), 0.3f), 1.0f);
    const float veg   = fmaf(0.5f, fminf(fmaxf(ndvv[j], 0.0f), 1.0f), 0.5f);
    const float base  = kPBurn * slope_eff * moist * veg;

    // cos(wm deg) = -sin(wd deg), sin(wm deg) = -cos(wd deg)
    // (mod 360 is invisible to sin/cos; cos(270-t) = -sin t, sin(270-t) = -cos t)
    const float wdr = wdv[j] * kDeg2Rad;
    const float cw = -__sinf(wdr);   // cos(wind_math)
    const float sw = -__cosf(wdr);   // sin(wind_math)
    const float u = kR2 * (cw + sw); // cos(45 - wm)
    const float v = kR2 * (cw - sw); // cos(315 - wm)
    const float wsc = kWWind * wsv[j];

    const int ci = lR * LDSW + lC;
    const float fsv = sF[ci];
    const float nN  = sF[ci - LDSW];
    const float nNE = sF[ci - LDSW + 1];
    const float nE  = sF[ci + 1];
    const float nSE = sF[ci + LDSW + 1];
    const float nS  = sF[ci + LDSW];
    const float nSW = sF[ci + LDSW - 1];
    const float nW  = sF[ci - 1];
    const float nNW = sF[ci - LDSW - 1];

    // clamp(base*(1+wsc*cd)) is monotone nondecreasing in cd, so the max over
    // burning directions equals the clamp applied at the max cosdiff.
    float m = (nN  > 0.5f) ?  sw : -4.0f;          // dir (-1, 0) @ 90
    m = fmaxf(m, (nNE > 0.5f) ?  u  : -4.0f);      // dir (-1, 1) @ 45
    m = fmaxf(m, (nE  > 0.5f) ?  cw : -4.0f);      // dir ( 0, 1) @ 0
    m = fmaxf(m, (nSE > 0.5f) ?  v  : -4.0f);      // dir ( 1, 1) @ 315
    m = fmaxf(m, (nS  > 0.5f) ? -sw : -4.0f);      // dir ( 1, 0) @ 270
    m = fmaxf(m, (nSW > 0.5f) ? -u  : -4.0f);      // dir ( 1,-1) @ 225
    m = fmaxf(m, (nW  > 0.5f) ? -cw : -4.0f);      // dir ( 0,-1) @ 180
    m = fmaxf(m, (nNW > 0.5f) ? -v  : -4.0f);      // dir (-1,-1) @ 135

    const float wf = fmaf(wsc, m, 1.0f);
    const float pv = fminf(fmaxf(base * wf, 0.0f), 1.0f);
    const float prob = (m > -3.0f) ? pv : 0.0f;    // no burning neighbor -> 0
    res[j] = fmaxf(fsv, (fsv < 0.5f) ? prob : 0.0f);
  }

  float4* o4 = (float4*)(out + (size_t)blockIdx.z * CS + grow);
  o4[0] = make_float4(res[0], res[1], res[2], res[3]);
  o4[1] = make_float4(res[4], res[5], res[6], res[7]);
}

extern "C" void kernel_launch(void* const* d_in, const int* in_sizes, int n_in,
                              void* d_out, int out_size, void* d_ws, size_t ws_size,
                              hipStream_t stream) {
  const float* x = (const float*)d_in[0];
  float* out = (float*)d_out;
  const int B = in_sizes[0] / (kC * kH * kW);   // 16
  dim3 grid(kW / TW, kH / TH, B);
  fire_spread_kernel<<<grid, NTHREADS, 0, stream>>>(x, out);
}